// SparseMHA_1357209665641
// MI455X (gfx1250) — compile-verified
//
#include <hip/hip_runtime.h>
#include <math.h>

#define HID 128
#define NHEAD 8
#define LDSPITCH 132   // 128 + 4 pad -> fragment ds_load_b64 bank-conflict-free

typedef __attribute__((ext_vector_type(2))) float v2f;
typedef __attribute__((ext_vector_type(4))) float v4f;
typedef __attribute__((ext_vector_type(8))) float v8f;

// -------- float atomic max via int/uint ordering trick --------
__device__ __forceinline__ void atomicMaxFloat(float* addr, float val) {
  if (val >= 0.0f)
    atomicMax((int*)addr, __float_as_int(val));
  else
    atomicMin((unsigned int*)addr, __float_as_uint(val));
}

// -------- Dense GEMM: Y[n,j] = (sum_c X[n,c]*W[j,c] + bias[j]) * scale --------
// Block = 256 threads = 8 waves. The block stages one 16x128 activation tile
// (shared by all 8 waves) and the full 128x128 weight matrix in LDS using
// coalesced global_load_b128, then each wave computes one 16x16 output tile
// with V_WMMA_F32_16X16X4_F32, fragments fed by conflict-free ds_load_b64.
__global__ __launch_bounds__(256)
void sgat_linear_wmma(const float* __restrict__ X, const float* __restrict__ W,
                      const float* __restrict__ B, float* __restrict__ Y,
                      float scale) {
  __shared__ float ldsA[16 * LDSPITCH];    //  ~8.3 KB
  __shared__ float ldsW[HID * LDSPITCH];   // ~66.0 KB

  const int tid     = threadIdx.x;
  const int lane    = tid & 31;
  const int wave    = tid >> 5;
  const int rowBase = blockIdx.x << 4;     // 16 nodes per block

  // ---- Stage A tile: 16 rows x 128 floats = 512 float4, 2 per thread ----
#pragma unroll
  for (int i = tid; i < 16 * 32; i += 256) {
    int r = i >> 5, qd = (i & 31) << 2;
    v4f val = *(const v4f*)(X + (size_t)(rowBase + r) * HID + qd);
    *(v4f*)(&ldsA[r * LDSPITCH + qd]) = val;
  }
  // ---- Stage W: 128 rows x 128 floats = 4096 float4, 16 per thread ----
#pragma unroll
  for (int i = tid; i < HID * 32; i += 256) {
    int r = i >> 5, qd = (i & 31) << 2;
    v4f val = *(const v4f*)(W + (size_t)r * HID + qd);
    *(v4f*)(&ldsW[r * LDSPITCH + qd]) = val;
  }
  __syncthreads();

  const int colBase = wave << 4;           // 16 output features per wave
  const int m       = lane & 15;
  const int kh      = (lane >> 4) << 1;    // K sub-offset 0 or 2 (A-frag layout)

  const float* ap = &ldsA[m * LDSPITCH + kh];
  const float* bp = &ldsW[(colBase + m) * LDSPITCH + kh];

  v8f acc = {};
#pragma unroll
  for (int k0 = 0; k0 < HID; k0 += 4) {
    v2f a = *(const v2f*)(ap + k0);
    v2f b = *(const v2f*)(bp + k0);
    acc = __builtin_amdgcn_wmma_f32_16x16x4_f32(false, a, false, b,
                                                (short)0, acc, false, false);
  }

  const int   nOut = colBase + m;
  const float bias = B[nOut];
  const int   rOff = rowBase + ((lane >> 4) << 3);  // D layout: M = r + 8*(lane/16)
#pragma unroll
  for (int r = 0; r < 8; ++r)
    Y[(size_t)(rOff + r) * HID + nOut] = (acc[r] + bias) * scale;
}

// -------- init per-(node,head) softmax state --------
__global__ void sgat_init_mz(float* __restrict__ m, float* __restrict__ z, int n) {
  int i = blockIdx.x * blockDim.x + threadIdx.x;
  if (i < n) { m[i] = -INFINITY; z[i] = 0.0f; }
}

__global__ void sgat_zero(float* __restrict__ p, int n) {
  int i = blockIdx.x * blockDim.x + threadIdx.x;
  if (i < n) p[i] = 0.0f;
}

// -------- per-(edge,head) score = <q[row], k[col]> over head_dim --------
// Feature layout from reshape(N, HEAD_DIM, NUM_HEADS): channel = d*8 + head.
__global__ __launch_bounds__(256)
void sgat_edge_scores(const float* __restrict__ q, const float* __restrict__ k,
                      const int* __restrict__ row, const int* __restrict__ col,
                      float* __restrict__ scores, float* __restrict__ mbuf, int E) {
  int idx = blockIdx.x * blockDim.x + threadIdx.x;
  if (idx >= E * NHEAD) return;
  int e = idx >> 3, hd = idx & 7;
  int r = row[e], c = col[e];
  const float* qp = q + (size_t)r * HID + hd;
  const float* kp = k + (size_t)c * HID + hd;
  float s = 0.0f;
#pragma unroll
  for (int d = 0; d < 16; ++d) s += qp[d * NHEAD] * kp[d * NHEAD];
  scores[idx] = s;
  atomicMaxFloat(&mbuf[r * NHEAD + hd], s);
}

// -------- e = exp(score - m[row]); z[row] += e (in-place on scores) --------
__global__ __launch_bounds__(256)
void sgat_edge_exp(float* __restrict__ ebuf, const float* __restrict__ mbuf,
                   const int* __restrict__ row, float* __restrict__ zbuf, int E) {
  int idx = blockIdx.x * blockDim.x + threadIdx.x;
  if (idx >= E * NHEAD) return;
  int e = idx >> 3, hd = idx & 7;
  int r = row[e];
  float ex = __expf(ebuf[idx] - mbuf[r * NHEAD + hd]);
  ebuf[idx] = ex;
  atomicAdd(&zbuf[r * NHEAD + hd], ex);
}

// -------- out[row, c] += (e/z) * v[col, c], one thread per (edge, channel) ----
__global__ __launch_bounds__(256)
void sgat_edge_aggr(const float* __restrict__ ebuf, const float* __restrict__ zbuf,
                    const float* __restrict__ v, const int* __restrict__ row,
                    const int* __restrict__ col, float* __restrict__ outp, int E) {
  int idx = blockIdx.x * blockDim.x + threadIdx.x;
  if (idx >= E * HID) return;
  int e = idx >> 7, c = idx & 127, hd = c & 7;
  int r = row[e], cl = col[e];
  float attn = ebuf[e * NHEAD + hd] / zbuf[r * NHEAD + hd];
  atomicAdd(&outp[(size_t)r * HID + c], attn * v[(size_t)cl * HID + c]);
}

extern "C" void kernel_launch(void* const* d_in, const int* in_sizes, int n_in,
                              void* d_out, int out_size, void* d_ws, size_t ws_size,
                              hipStream_t stream) {
  const float* h     = (const float*)d_in[0];
  const int*   row   = (const int*)d_in[1];
  const int*   col   = (const int*)d_in[2];
  const float* q_w   = (const float*)d_in[3];
  const float* q_b   = (const float*)d_in[4];
  const float* k_w   = (const float*)d_in[5];
  const float* k_b   = (const float*)d_in[6];
  const float* v_w   = (const float*)d_in[7];
  const float* v_b   = (const float*)d_in[8];
  const float* out_w = (const float*)d_in[9];
  const float* out_b = (const float*)d_in[10];
  float*       out   = (float*)d_out;

  const int N = in_sizes[0] / HID;   // 50000 (multiple of 16)
  const int E = in_sizes[1];         // 800000

  // Workspace layout (floats). outp reuses the q buffer once scores are done.
  float* q    = (float*)d_ws;                 // N*128
  float* k    = q    + (size_t)N * HID;       // N*128
  float* v    = k    + (size_t)N * HID;       // N*128
  float* ebuf = v    + (size_t)N * HID;       // E*8
  float* mbuf = ebuf + (size_t)E * NHEAD;     // N*8
  float* zbuf = mbuf + (size_t)N * NHEAD;     // N*8
  float* outp = q;                            // reuse q after scores

  const float scaling = 0.25f;  // head_dim^-0.5 = 16^-0.5
  const int rowTiles  = N / 16;

  // 1) Projections (WMMA f32 GEMMs, LDS-staged tiles)
  sgat_linear_wmma<<<rowTiles, 256, 0, stream>>>(h, q_w, q_b, q, scaling);
  sgat_linear_wmma<<<rowTiles, 256, 0, stream>>>(h, k_w, k_b, k, 1.0f);
  sgat_linear_wmma<<<rowTiles, 256, 0, stream>>>(h, v_w, v_b, v, 1.0f);

  // 2) Softmax state init
  int nmz = N * NHEAD;
  sgat_init_mz<<<(nmz + 255) / 256, 256, 0, stream>>>(mbuf, zbuf, nmz);

  // 3) Edge scores + running per-(node,head) max
  int nEH = E * NHEAD;
  sgat_edge_scores<<<(nEH + 255) / 256, 256, 0, stream>>>(q, k, row, col, ebuf, mbuf, E);

  // 4) Zero the aggregation accumulator (q buffer now free)
  int nNC = N * HID;
  sgat_zero<<<(nNC + 255) / 256, 256, 0, stream>>>(outp, nNC);

  // 5) exp + denominator
  sgat_edge_exp<<<(nEH + 255) / 256, 256, 0, stream>>>(ebuf, mbuf, row, zbuf, E);

  // 6) Weighted aggregation of v into outp
  int nEC = E * HID;
  sgat_edge_aggr<<<(nEC + 255) / 256, 256, 0, stream>>>(ebuf, zbuf, v, row, col, outp, E);

  // 7) Output projection (WMMA f32 GEMM) straight into d_out
  sgat_linear_wmma<<<rowTiles, 256, 0, stream>>>(outp, out_w, out_b, out, 1.0f);
}